// EntRelJointDecoder_89867895701995
// MI455X (gfx1250) — compile-verified
//
#include <hip/hip_runtime.h>
#include <hip/hip_bf16.h>
#include <math.h>

// ---------------------------------------------------------------------------
// EntRelJointDecoder fused forward for MI455X (gfx1250, wave32, WMMA)
//   B=2, S=112, H=768, M=256, L=54, O=10
//   Dominant GEMM: pair[12544x256] x vU^T[256x1120] per batch (f32, 14.4 GFLOP)
//   - V_WMMA_F32_16X16X4_F32 (exact fp32) for every contraction
//   - double-buffered LDS vU tiles streamed with GLOBAL_LOAD_ASYNC_TO_LDS_B128
//     (ASYNCcnt) when the toolchain exposes the builtin; plain-load fallback.
// ---------------------------------------------------------------------------

typedef __attribute__((ext_vector_type(2))) float v2f;
typedef __attribute__((ext_vector_type(8))) float v8f;

#define Bc   2
#define Ss   112
#define Hh   768
#define Mm   256
#define Ll   54
#define Oo   10
#define XY   (Ss*Ss)      // 12544 rows of pair per batch
#define ZO   (Ss*Oo)      // 1120 cols of q_score per batch
#define RBS  (Bc*Ss)      // 224 rows for the [B*S, ...] projections
#define LDA  260          // padded LDS row pitch (floats): 260%64=4 -> no bank conflicts

#if __has_builtin(__builtin_amdgcn_global_load_async_to_lds_b128) && \
    __has_builtin(__builtin_amdgcn_s_wait_asynccnt)
#define HAVE_ASYNC 1
// builtin signature (from probe): (v4i addrspace(1)*, v4i addrspace(3)*, Imm, Imm)
typedef __attribute__((ext_vector_type(4))) int v4i;
typedef __attribute__((address_space(1))) v4i* g128_t;
typedef __attribute__((address_space(3))) v4i* l128_t;
#else
#define HAVE_ASYNC 0
#endif

__device__ __forceinline__ float gelu_f(float v) {
    return 0.5f * v * (1.0f + erff(v * 0.70710678118654752f));
}

__device__ __forceinline__ v8f wmma16x16x4(v2f a, v2f b, v8f c) {
    // D = A(16x4 f32) * B(4x16 f32) + C(16x16 f32), exact fp32
    return __builtin_amdgcn_wmma_f32_16x16x4_f32(false, a, false, b, (short)0, c,
                                                 false, false);
}

// ---------------------------------------------------------------------------
// C[224 x 256] = A[224 x K] @ W[K x 256]  (+bias, +gelu) ; one wave per tile
// ---------------------------------------------------------------------------
__global__ void __launch_bounds__(32)
proj_gemm_kernel(const float* __restrict__ A, const float* __restrict__ W,
                 const float* __restrict__ bias, float* __restrict__ C,
                 int K, int do_gelu) {
    const int rt = blockIdx.x, ct = blockIdx.y;
    const int lane = threadIdx.x;
    const int l15 = lane & 15, g = lane >> 4;
    const int row = rt * 16 + l15;
    const int col = ct * 16 + l15;
    v8f acc = {};
    for (int k0 = 0; k0 < K; k0 += 4) {
        const int ka = k0 + 2 * g;           // lanes 0-15: K=k0,k0+1 ; 16-31: K=k0+2,k0+3
        v2f a, b;
        a.x = A[row * K + ka];      a.y = A[row * K + ka + 1];
        b.x = W[ka * Mm + col];     b.y = W[(ka + 1) * Mm + col];
        acc = wmma16x16x4(a, b, acc);
    }
    #pragma unroll
    for (int v = 0; v < 8; ++v) {
        const int r = rt * 16 + v + 8 * g;   // C/D layout: M = vgpr + 8*(lane/16)
        float val = acc[v];
        if (bias)    val += bias[col];
        if (do_gelu) val = gelu_f(val);
        C[r * Mm + col] = val;
    }
}

// ---------------------------------------------------------------------------
// vU[bz, o, i] = sum_j value[bz, j] * U[o, i, j] ; one wave per 16x16 tile
// ---------------------------------------------------------------------------
__global__ void __launch_bounds__(32)
vu_gemm_kernel(const float* __restrict__ value, const float* __restrict__ U,
               float* __restrict__ vU) {
    const int rt = blockIdx.x, ct = blockIdx.y, o = blockIdx.z;
    const int lane = threadIdx.x;
    const int l15 = lane & 15, g = lane >> 4;
    const int row = rt * 16 + l15;           // bz
    const int col = ct * 16 + l15;           // i
    const float* Uo = U + (size_t)o * Mm * Mm;
    v8f acc = {};
    for (int k0 = 0; k0 < Mm; k0 += 4) {
        const int ka = k0 + 2 * g;
        v2f a, b;
        a.x = value[row * Mm + ka]; a.y = value[row * Mm + ka + 1];
        b.x = Uo[col * Mm + ka];    b.y = Uo[col * Mm + ka + 1];  // B[k][n]=U[o][n][k]
        acc = wmma16x16x4(a, b, acc);
    }
    #pragma unroll
    for (int v = 0; v < 8; ++v) {
        const int r = rt * 16 + v + 8 * g;
        vU[((size_t)r * Oo + o) * Mm + col] = acc[v];
    }
}

// ---------------------------------------------------------------------------
// Fused hot path: per block (8 waves) -> 32 rows of pair (recomputed with gelu
// into LDS), swept against all 1120 vU columns (q_score) and final_w (joint).
// vU tiles are double-buffered: tile s+1 streams into LDS (async when
// available) while tile s feeds the WMMA pipe.
// ---------------------------------------------------------------------------
__global__ void __launch_bounds__(256)
qjoint_kernel(const float* __restrict__ hpart, const float* __restrict__ tpart,
              const float* __restrict__ pair_b, const float* __restrict__ vU,
              const float* __restrict__ final_w, const float* __restrict__ final_b,
              float* __restrict__ qout, float* __restrict__ jout) {
    extern __shared__ float smem[];
    float* As  = smem;                   // [32][LDA] pair tile
    float* Bs0 = smem + 32 * LDA;        // [64][LDA] vU tile, buffer 0
    float* Bs1 = Bs0 + 64 * LDA;         // [64][LDA] vU tile, buffer 1

    const int blk = blockIdx.x;
    const int b  = blk / (XY / 32);
    const int rt = blk % (XY / 32);
    const int tid = threadIdx.x;
    const float* vUb = vU + (size_t)b * ZO * Mm;
    float* qb = qout + (size_t)b * XY * ZO;

    // ---- stage vU columns [s*64, s*64+64) into buf ------------------------
    auto stage_B = [&](int s, float* buf) {
#if HAVE_ASYNC
        // 64 rows x 1KB, 16B per lane-transfer: memory -> LDS, no VGPR data
        for (int idx = tid; idx < 64 * 64; idx += 256) {
            const int cL = idx >> 6;                 // local column 0..63
            const int xo = (idx & 63) * 4;           // float offset within row
            const int c  = s * 64 + cL;
            if (c < ZO) {
                __builtin_amdgcn_global_load_async_to_lds_b128(
                    (g128_t)(vUb + (size_t)c * Mm + xo),
                    (l128_t)(buf + cL * LDA + xo), 0, 0);
            }
        }
#else
        for (int e = tid; e < 64 * Mm; e += 256) {
            const int cL = e >> 8, k = e & 255;
            const int c = s * 64 + cL;
            if (c < ZO) buf[cL * LDA + k] = vUb[(size_t)c * Mm + k];
        }
        {   // prefetch the tile after next (global_prefetch_b8)
            const int nc = (s + 1) * 64 + (tid & 63);
            if (nc < ZO) __builtin_prefetch(&vUb[(size_t)nc * Mm], 0, 1);
        }
#endif
    };

    // ---- stage A: pair[r][k] = gelu(h[x][k] + t[y][k] + pb[k]) -------------
    for (int e = tid; e < 32 * Mm; e += 256) {
        const int i = e >> 8, k = e & 255;
        const int r = rt * 32 + i;
        const int x = r / Ss, y = r % Ss;
        const float v = hpart[(b * Ss + x) * Mm + k]
                      + tpart[(b * Ss + y) * Mm + k] + pair_b[k];
        As[i * LDA + k] = gelu_f(v);
    }
    stage_B(0, Bs0);                      // prologue fill of buffer 0

    const int wave = tid >> 5, lane = tid & 31;
    const int l15 = lane & 15, g = lane >> 4;
    const int wr = (wave >> 2) * 16;      // 0 / 16 : row half
    const int wc = wave & 3;              // 0..3   : col sub-tile

    // ---- q_score: 18 steps of 64 columns (70 tiles of 16; ZO=1120) ---------
    for (int s = 0; s < 18; ++s) {
        float* Bcur = (s & 1) ? Bs1 : Bs0;
        float* Bnxt = (s & 1) ? Bs0 : Bs1;
#if HAVE_ASYNC
        __builtin_amdgcn_s_wait_asynccnt(0);   // tile s landed in LDS
#endif
        __syncthreads();                        // visible to all; prev compute done
        if (s + 1 < 18) stage_B(s + 1, Bnxt);   // stream next tile under compute

        const int ct = s * 4 + wc;              // global 16-col tile index
        if (ct < ZO / 16) {                     // wave-uniform: EXEC all-1s at WMMA
            const float* Ap = As   + (wr + l15) * LDA;
            const float* Bp = Bcur + (wc * 16 + l15) * LDA;
            v8f acc = {};
            #pragma unroll 8
            for (int k0 = 0; k0 < Mm; k0 += 4) {
                const int ka = k0 + 2 * g;
                v2f a, bb;
                a.x  = Ap[ka]; a.y  = Ap[ka + 1];
                bb.x = Bp[ka]; bb.y = Bp[ka + 1];
                acc = wmma16x16x4(a, bb, acc);
            }
            const int cg = ct * 16 + l15;
            #pragma unroll
            for (int v = 0; v < 8; ++v) {
                const int rloc = wr + v + 8 * g;
                qb[(size_t)(rt * 32 + rloc) * ZO + cg] = acc[v];
            }
        }
    }

    // ---- joint_score: same pair tile vs final_w^T (54 cols, padded to 64) --
#if HAVE_ASYNC
    __builtin_amdgcn_s_wait_asynccnt(0);
#endif
    // Bs0 was last read in step 16 (finished before step 17's barrier); safe.
    for (int e = tid; e < 64 * Mm; e += 256) {
        const int cL = e >> 8, k = e & 255;
        Bs0[cL * LDA + k] = (cL < Ll) ? final_w[k * Ll + cL] : 0.0f;
    }
    __syncthreads();
    {
        const float* Ap = As  + (wr + l15) * LDA;
        const float* Bp = Bs0 + (wc * 16 + l15) * LDA;
        v8f acc = {};
        #pragma unroll 8
        for (int k0 = 0; k0 < Mm; k0 += 4) {
            const int ka = k0 + 2 * g;
            v2f a, bb;
            a.x  = Ap[ka]; a.y  = Ap[ka + 1];
            bb.x = Bp[ka]; bb.y = Bp[ka + 1];
            acc = wmma16x16x4(a, bb, acc);
        }
        const int col = wc * 16 + l15;
        if (col < Ll) {
            const float bias = final_b[col];
            #pragma unroll
            for (int v = 0; v < 8; ++v) {
                const int rloc = wr + v + 8 * g;
                jout[((size_t)b * XY + rt * 32 + rloc) * Ll + col] = acc[v] + bias;
            }
        }
    }
}

// ---------------------------------------------------------------------------
// Losses
// ---------------------------------------------------------------------------
__global__ void init_acc_kernel(float* acc) {
    if (threadIdx.x < 4) acc[threadIdx.x] = 0.0f;
}

__global__ void __launch_bounds__(256)
element_loss_kernel(const float* __restrict__ jout, const int* __restrict__ labels,
                    const int* __restrict__ mask, float* __restrict__ acc) {
    const int p = blockIdx.x * 256 + threadIdx.x;
    float nll = 0.0f, cnt = 0.0f;
    if (p < Bc * XY && mask[p] != 0) {
        const float* lg = jout + (size_t)p * Ll;
        float mx = -1e30f;
        for (int i = 0; i < Ll; ++i) mx = fmaxf(mx, lg[i]);
        float se = 0.0f;
        for (int i = 0; i < Ll; ++i) se += expf(lg[i] - mx);
        nll = (logf(se) + mx) - lg[labels[p]];
        cnt = 1.0f;
    }
    __shared__ float s1[256], s2[256];
    s1[threadIdx.x] = nll; s2[threadIdx.x] = cnt;
    __syncthreads();
    for (int st = 128; st > 0; st >>= 1) {
        if (threadIdx.x < st) {
            s1[threadIdx.x] += s1[threadIdx.x + st];
            s2[threadIdx.x] += s2[threadIdx.x + st];
        }
        __syncthreads();
    }
    if (threadIdx.x == 0) { atomicAdd(&acc[0], s1[0]); atomicAdd(&acc[1], s2[0]); }
}

__global__ void __launch_bounds__(256)
q_loss_kernel(const float* __restrict__ qout, const int* __restrict__ labels,
              const int* __restrict__ mask, float* __restrict__ acc) {
    const int p = blockIdx.x * 256 + threadIdx.x;
    float nll = 0.0f, cnt = 0.0f;
    if (p < Bc * XY * Ss && mask[p] != 0) {
        const float* lg = qout + (size_t)p * Oo;
        // reference quirk: CE applied to softmax(q_score)
        float mx = -1e30f;
        for (int i = 0; i < Oo; ++i) mx = fmaxf(mx, lg[i]);
        float se = 0.0f;
        float pr[Oo];
        for (int i = 0; i < Oo; ++i) { pr[i] = expf(lg[i] - mx); se += pr[i]; }
        const float inv = 1.0f / se;
        float mx2 = -1e30f;
        for (int i = 0; i < Oo; ++i) { pr[i] *= inv; mx2 = fmaxf(mx2, pr[i]); }
        float se2 = 0.0f;
        for (int i = 0; i < Oo; ++i) se2 += expf(pr[i] - mx2);
        nll = (logf(se2) + mx2) - pr[labels[p]];
        cnt = 1.0f;
    }
    __shared__ float s1[256], s2[256];
    s1[threadIdx.x] = nll; s2[threadIdx.x] = cnt;
    __syncthreads();
    for (int st = 128; st > 0; st >>= 1) {
        if (threadIdx.x < st) {
            s1[threadIdx.x] += s1[threadIdx.x + st];
            s2[threadIdx.x] += s2[threadIdx.x + st];
        }
        __syncthreads();
    }
    if (threadIdx.x == 0) { atomicAdd(&acc[2], s1[0]); atomicAdd(&acc[3], s2[0]); }
}

__global__ void finalize_kernel(const float* __restrict__ acc, float* __restrict__ out) {
    out[0] = acc[0] / fmaxf(acc[1], 1.0f) + acc[2] / fmaxf(acc[3], 1.0f);
}

// ---------------------------------------------------------------------------
extern "C" void kernel_launch(void* const* d_in, const int* in_sizes, int n_in,
                              void* d_out, int out_size, void* d_ws, size_t ws_size,
                              hipStream_t stream) {
    (void)in_sizes; (void)n_in; (void)out_size; (void)ws_size;
    const float* x       = (const float*)d_in[0];
    const int*   jlab    = (const int*)d_in[1];
    const int*   jmask   = (const int*)d_in[2];
    const int*   qlab    = (const int*)d_in[3];
    const int*   qmask   = (const int*)d_in[4];
    const float* pair_w  = (const float*)d_in[5];
    const float* pair_b  = (const float*)d_in[6];
    const float* final_w = (const float*)d_in[7];
    const float* final_b = (const float*)d_in[8];
    const float* value_w = (const float*)d_in[9];
    const float* value_b = (const float*)d_in[10];
    const float* U       = (const float*)d_in[11];

    // workspace: h[224*256] t[224*256] value[224*256] vU[224*10*256] acc[4] (~3 MB)
    float* ws    = (float*)d_ws;
    float* hpart = ws;
    float* tpart = hpart + RBS * Mm;
    float* value = tpart + RBS * Mm;
    float* vU    = value + RBS * Mm;
    float* acc   = vU + (size_t)RBS * Oo * Mm;

    float* out  = (float*)d_out;
    float* loss = out;
    float* jout = out + 1;
    float* qout = jout + (size_t)Bc * XY * Ll;

    const dim3 g1(RBS / 16, Mm / 16);
    proj_gemm_kernel<<<g1, 32, 0, stream>>>(x, pair_w,            nullptr, hpart, Hh, 0);
    proj_gemm_kernel<<<g1, 32, 0, stream>>>(x, pair_w + Hh * Mm,  nullptr, tpart, Hh, 0);
    proj_gemm_kernel<<<g1, 32, 0, stream>>>(x, value_w,           value_b, value, Hh, 1);

    const dim3 g2(RBS / 16, Mm / 16, Oo);
    vu_gemm_kernel<<<g2, 32, 0, stream>>>(value, U, vU);

    // LDS: A tile + 2 x B tile (double buffer) = 166,400 B (< 320 KB/WG)
    const size_t lds_bytes = (size_t)(32 * LDA + 2 * 64 * LDA) * sizeof(float);
    qjoint_kernel<<<Bc * (XY / 32), 256, lds_bytes, stream>>>(
        hpart, tpart, pair_b, vU, final_w, final_b, qout, jout);

    init_acc_kernel<<<1, 32, 0, stream>>>(acc);
    element_loss_kernel<<<(Bc * XY + 255) / 256, 256, 0, stream>>>(jout, jlab, jmask, acc);
    q_loss_kernel<<<(Bc * XY * Ss + 255) / 256, 256, 0, stream>>>(qout, qlab, qmask, acc);
    finalize_kernel<<<1, 1, 0, stream>>>(acc, loss);
}